// Mamba2Block_91225105367674
// MI455X (gfx1250) — compile-verified
//
#include <hip/hip_runtime.h>
#include <hip/hip_bf16.h>
#include <math.h>

// ---------------------------------------------------------------------------
// Mamba2 block forward for MI455X (gfx1250, wave32).
// GEMMs use full-precision V_WMMA_F32_16X16X4_F32 with 2x2 register blocking
// (32x32 output per wave): each A/B fragment feeds two WMMAs, so the inner
// loop runs at 1 fragment-load per WMMA instead of 2 — the GEMMs are
// L2-load-issue bound at this size, not WMMA bound.
// The sequential scan prefetches the next timestep's B/C row
// (global_prefetch_b8) to hide load latency behind the 64-FMA chain.
// ---------------------------------------------------------------------------

typedef __attribute__((ext_vector_type(2))) float v2f;
typedef __attribute__((ext_vector_type(8))) float v8f;

#define NB       2
#define LSEQ     1024
#define DMODEL   512
#define DSTATE   128
#define DCONVW   4
#define HEADDIM  64
#define DINNER   1024
#define NHEADS   16
#define CONVDIM  1280   // DINNER + 2*DSTATE
#define DPROJ    2320   // 2*DINNER + 2*DSTATE + NHEADS
#define MROWS    (NB*LSEQ)  // 2048

__device__ __forceinline__ float silu_f(float v) { return v / (1.0f + expf(-v)); }

// ---------------------------------------------------------------------------
// fp32 WMMA GEMM:  C[m,n] = sum_k A[m,k] * W[n,k]
// A: (M x K) row-major, lda = K.  W: (N x K) row-major, ldw = K (acts as B^T).
// One wave computes a 32x32 C block (2x2 WMMA tiles, 4 accumulators).
// A-frag layout (ISA 7.12.2): lane&15 = row M, VGPR0/1 = K + 2*(lane>>4) +0/+1.
// B-frag mirrored with lane&15 = col N. C/D: VGPR r -> row r + 8*(lane>>4).
// M must be a multiple of 32; odd ntiles handled by computing the second
// N-tile against replicated data and skipping its store (keeps the K-loop
// uniform so EXEC is all-1s for every WMMA).
// ---------------------------------------------------------------------------
__global__ void __launch_bounds__(128)
wmma_gemm_f32(const float* __restrict__ A, const float* __restrict__ W,
              float* __restrict__ C, int K, int lda, int ldw, int ldc, int ntiles)
{
    const int lane = threadIdx.x & 31;
    const int wave = threadIdx.x >> 5;
    const int ntp  = blockIdx.y * 4 + wave;      // n tile-pair index
    const int nt0  = ntp * 2;
    const int nt1  = nt0 + 1;
    if (nt0 >= ntiles) return;                   // wave-uniform guard
    const bool hasN1 = (nt1 < ntiles);

    const int m0   = blockIdx.x * 32;
    const int half = lane >> 4;                  // 0 or 1
    const int l16  = lane & 15;

    const float* __restrict__ a0row = A + (size_t)(m0 + l16) * lda;
    const float* __restrict__ a1row = a0row + (size_t)16 * lda;
    const float* __restrict__ w0row = W + (size_t)(nt0 * 16 + l16) * ldw;
    const float* __restrict__ w1row = hasN1 ? (W + (size_t)(nt1 * 16 + l16) * ldw)
                                            : w0row;   // dummy, result discarded

    v8f acc00 = {}, acc01 = {}, acc10 = {}, acc11 = {};
    for (int k0 = 0; k0 < K; k0 += 4) {
        const int kb = k0 + 2 * half;
        v2f a0, a1, b0, b1;
        a0.x = a0row[kb]; a0.y = a0row[kb + 1];
        a1.x = a1row[kb]; a1.y = a1row[kb + 1];
        b0.x = w0row[kb]; b0.y = w0row[kb + 1];
        b1.x = w1row[kb]; b1.y = w1row[kb + 1];
        acc00 = __builtin_amdgcn_wmma_f32_16x16x4_f32(false, a0, false, b0, (short)0, acc00, false, false);
        acc10 = __builtin_amdgcn_wmma_f32_16x16x4_f32(false, a1, false, b0, (short)0, acc10, false, false);
        acc01 = __builtin_amdgcn_wmma_f32_16x16x4_f32(false, a0, false, b1, (short)0, acc01, false, false);
        acc11 = __builtin_amdgcn_wmma_f32_16x16x4_f32(false, a1, false, b1, (short)0, acc11, false, false);
    }

    float* __restrict__ c0 = C + (size_t)(m0 + 8 * half) * ldc + nt0 * 16 + l16;
    float* __restrict__ c1 = c0 + (size_t)16 * ldc;
#pragma unroll
    for (int r = 0; r < 8; ++r) c0[(size_t)r * ldc] = acc00[r];
#pragma unroll
    for (int r = 0; r < 8; ++r) c1[(size_t)r * ldc] = acc10[r];
    if (hasN1) {
#pragma unroll
        for (int r = 0; r < 8; ++r) c0[(size_t)r * ldc + 16] = acc01[r];
#pragma unroll
        for (int r = 0; r < 8; ++r) c1[(size_t)r * ldc + 16] = acc11[r];
    }
}

// ---------------------------------------------------------------------------
// Depthwise causal conv (width 4) + bias + SiLU over the xBC slice of zxbcdt.
// ---------------------------------------------------------------------------
__global__ void __launch_bounds__(256)
conv_silu_kernel(const float* __restrict__ zx, const float* __restrict__ conv_w,
                 const float* __restrict__ conv_b, float* __restrict__ xbc)
{
    const int e = blockIdx.x * blockDim.x + threadIdx.x;
    if (e >= MROWS * CONVDIM) return;
    const int bl = e / CONVDIM;
    const int c  = e % CONVDIM;
    const int l  = bl % LSEQ;
    float s = conv_b[c];
#pragma unroll
    for (int j = 0; j < DCONVW; ++j) {
        const int lj = l - (DCONVW - 1) + j;
        if (lj >= 0)
            s += zx[(size_t)(bl + lj - l) * DPROJ + DINNER + c] * conv_w[c * DCONVW + j];
    }
    xbc[e] = silu_f(s);
}

// ---------------------------------------------------------------------------
// dt = softplus(raw + dt_bias); dA = exp(dt * (-exp(A_log)))
// ---------------------------------------------------------------------------
__global__ void __launch_bounds__(256)
dt_kernel(const float* __restrict__ zx, const float* __restrict__ dt_bias,
          const float* __restrict__ A_log, float* __restrict__ dtb, float* __restrict__ dab)
{
    const int e = blockIdx.x * blockDim.x + threadIdx.x;
    if (e >= MROWS * NHEADS) return;
    const int bl = e / NHEADS;
    const int h  = e % NHEADS;
    const float raw = zx[(size_t)bl * DPROJ + DINNER + CONVDIM + h] + dt_bias[h];
    const float dt  = (raw > 20.0f) ? raw : log1pf(expf(raw));
    const float Ah  = -expf(A_log[h]);
    dtb[e] = dt;
    dab[e] = expf(dt * Ah);
}

// ---------------------------------------------------------------------------
// Sequential SSM scan. One block per (batch, head); 256 threads.
// Thread (p = tid>>2, j = tid&3) keeps h[p, j*32 .. j*32+31] in 32 VGPRs.
// B/C rows (256 floats) staged in LDS each timestep; next row prefetched
// (global_prefetch_b8) to overlap latency with the FMA chain; y reduced via
// shfl_xor across the 4 lanes sharing p (wave32).
// ---------------------------------------------------------------------------
__global__ void __launch_bounds__(256)
scan_kernel(const float* __restrict__ xbc, const float* __restrict__ dtb,
            const float* __restrict__ dab, const float* __restrict__ Dp,
            float* __restrict__ y)
{
    __shared__ float sBC[2 * DSTATE];   // [0,128) = B row, [128,256) = C row
    const int b   = blockIdx.x / NHEADS;
    const int hd  = blockIdx.x % NHEADS;
    const int tid = threadIdx.x;
    const int p   = tid >> 2;
    const int j   = tid & 3;
    const int nb  = j * 32;
    const float Dh = Dp[hd];

    float h[32];
#pragma unroll
    for (int i = 0; i < 32; ++i) h[i] = 0.0f;

    for (int t = 0; t < LSEQ; ++t) {
        const int row = b * LSEQ + t;
        const float* __restrict__ xrow = xbc + (size_t)row * CONVDIM;
        sBC[tid] = xrow[DINNER + tid];          // 256 threads load B(128)+C(128)
        if (t + 1 < LSEQ)                       // hide next step's load latency
            __builtin_prefetch(xrow + CONVDIM + DINNER + tid, 0, 3);
        __syncthreads();

        const float dAv = dab[row * NHEADS + hd];
        const float dtv = dtb[row * NHEADS + hd];
        const float xv  = xrow[hd * HEADDIM + p];
        const float dxv = dtv * xv;

        float accv = 0.0f;
#pragma unroll
        for (int i = 0; i < 32; ++i) {
            h[i] = fmaf(h[i], dAv, dxv * sBC[nb + i]);
            accv = fmaf(h[i], sBC[DSTATE + nb + i], accv);
        }
        accv += __shfl_xor(accv, 1, 32);
        accv += __shfl_xor(accv, 2, 32);
        if (j == 0)
            y[(size_t)row * DINNER + hd * HEADDIM + p] = accv + Dh * xv;
        __syncthreads();
    }
}

// ---------------------------------------------------------------------------
// g = y * silu(z); g *= rsqrt(mean(g^2) + eps) * norm_w.  In place on y.
// One block (256 threads) per row of 1024 elements.
// ---------------------------------------------------------------------------
__global__ void __launch_bounds__(256)
gate_rms_kernel(const float* __restrict__ zx, const float* __restrict__ norm_w,
                float* __restrict__ y)
{
    __shared__ float red[8];
    const int row = blockIdx.x;
    const int tid = threadIdx.x;

    float g[4];
    float ss = 0.0f;
#pragma unroll
    for (int i = 0; i < 4; ++i) {
        const int k  = tid + i * 256;
        const float zv = zx[(size_t)row * DPROJ + k];
        const float yv = y[(size_t)row * DINNER + k];
        const float gv = yv * silu_f(zv);
        g[i] = gv;
        ss += gv * gv;
    }
#pragma unroll
    for (int off = 16; off >= 1; off >>= 1) ss += __shfl_xor(ss, off, 32);
    if ((tid & 31) == 0) red[tid >> 5] = ss;
    __syncthreads();
    if (tid < 8) {
        float v = red[tid];
        v += __shfl_xor(v, 4, 8);
        v += __shfl_xor(v, 2, 8);
        v += __shfl_xor(v, 1, 8);
        if (tid == 0) red[0] = v;
    }
    __syncthreads();
    const float rms = rsqrtf(red[0] / (float)DINNER + 1e-5f);
#pragma unroll
    for (int i = 0; i < 4; ++i) {
        const int k = tid + i * 256;
        y[(size_t)row * DINNER + k] = g[i] * rms * norm_w[k];
    }
}

// ---------------------------------------------------------------------------
// Launch: ws layout (floats):
//   zx  : MROWS*DPROJ   (19.0 MB)
//   xbc : MROWS*CONVDIM (10.5 MB)
//   dtb : MROWS*NHEADS
//   dab : MROWS*NHEADS
//   y   : MROWS*DINNER  ( 8.4 MB)   total ~38.2 MB
// ---------------------------------------------------------------------------
extern "C" void kernel_launch(void* const* d_in, const int* in_sizes, int n_in,
                              void* d_out, int out_size, void* d_ws, size_t ws_size,
                              hipStream_t stream)
{
    const float* x        = (const float*)d_in[0];
    const float* in_proj  = (const float*)d_in[1];
    const float* conv_w   = (const float*)d_in[2];
    const float* conv_b   = (const float*)d_in[3];
    const float* dt_bias  = (const float*)d_in[4];
    const float* A_log    = (const float*)d_in[5];
    const float* Dp       = (const float*)d_in[6];
    const float* norm_w   = (const float*)d_in[7];
    const float* out_proj = (const float*)d_in[8];
    (void)in_sizes; (void)n_in; (void)out_size; (void)ws_size;

    float* ws  = (float*)d_ws;
    float* zx  = ws;
    float* xbc = zx  + (size_t)MROWS * DPROJ;
    float* dtb = xbc + (size_t)MROWS * CONVDIM;
    float* dab = dtb + (size_t)MROWS * NHEADS;
    float* y   = dab + (size_t)MROWS * NHEADS;
    float* out = (float*)d_out;

    // 1) in_proj: (2048 x 512) @ (2320 x 512)^T -> zx (2048 x 2320)
    {
        const int ntiles = DPROJ / 16;                 // 145 (odd -> edge pair)
        const int npairs = (ntiles + 1) / 2;           // 73
        dim3 grid(MROWS / 32, (npairs + 3) / 4);       // 64 x 19
        wmma_gemm_f32<<<grid, 128, 0, stream>>>(x, in_proj, zx,
                                                DMODEL, DMODEL, DMODEL, DPROJ, ntiles);
    }
    // 2) depthwise conv + SiLU
    conv_silu_kernel<<<(MROWS * CONVDIM + 255) / 256, 256, 0, stream>>>(zx, conv_w, conv_b, xbc);
    // 3) dt / dA precompute
    dt_kernel<<<(MROWS * NHEADS + 255) / 256, 256, 0, stream>>>(zx, dt_bias, A_log, dtb, dab);
    // 4) sequential scan -> y
    scan_kernel<<<NB * NHEADS, 256, 0, stream>>>(xbc, dtb, dab, Dp, y);
    // 5) gate + RMSNorm (in place on y)
    gate_rms_kernel<<<MROWS, 256, 0, stream>>>(zx, norm_w, y);
    // 6) out_proj: (2048 x 1024) @ (512 x 1024)^T -> d_out (2048 x 512)
    {
        const int ntiles = DMODEL / 16;                // 32 -> 16 pairs
        const int npairs = ntiles / 2;
        dim3 grid(MROWS / 32, npairs / 4);             // 64 x 4
        wmma_gemm_f32<<<grid, 128, 0, stream>>>(y, out_proj, out,
                                                DINNER, DINNER, DINNER, DMODEL, ntiles);
    }
}